// BoxContrastHead_16793322127473
// MI455X (gfx1250) — compile-verified
//
#include <hip/hip_runtime.h>
#include <hip/hip_bf16.h>

// Problem constants (match reference)
#define BB   8
#define GG   256
#define CC   256
#define HH   188
#define WWID 188
#define FF1  256
#define N2   2048            // rows of z_interval / z_box
#define NNEG 2046

typedef __attribute__((ext_vector_type(2))) float v2f;
typedef __attribute__((ext_vector_type(8))) float v8f;

// ---------------------------------------------------------------------------
// K1: bilinear gather + pair interleave -> z_interval [2048, 256]
// one block per (b,g) box, threads over channels
// ---------------------------------------------------------------------------
__global__ void k_gather(const float* __restrict__ bev,
                         const float* __restrict__ centers,
                         float* __restrict__ zi) {
  int bg = blockIdx.x;
  int b  = bg >> 8;            // / GG
  int g  = bg & 255;           // % GG
  int r  = (((b >> 1) << 8) + g) * 2 + (b & 1);  // interleaved row
  int c  = threadIdx.x;

  const float* ctr = centers + ((size_t)b * GG + g) * 3;
  float x = (ctr[0] - (-75.2f)) / (0.1f * 8.0f);
  float y = (ctr[1] - (-75.2f)) / (0.1f * 8.0f);

  float x0f = floorf(x), y0f = floorf(y);
  int x0 = (int)x0f;     x0 = x0 < 0 ? 0 : (x0 > WWID - 1 ? WWID - 1 : x0);
  int x1 = (int)x0f + 1; x1 = x1 < 0 ? 0 : (x1 > WWID - 1 ? WWID - 1 : x1);
  int y0 = (int)y0f;     y0 = y0 < 0 ? 0 : (y0 > HH - 1 ? HH - 1 : y0);
  int y1 = (int)y0f + 1; y1 = y1 < 0 ? 0 : (y1 > HH - 1 ? HH - 1 : y1);

  float x0c = (float)x0, x1c = (float)x1, y0c = (float)y0, y1c = (float)y1;
  float wa = (x1c - x) * (y1c - y);
  float wb = (x1c - x) * (y - y0c);
  float wc = (x - x0c) * (y1c - y);
  float wd = (x - x0c) * (y - y0c);

  const float* base = bev + ((size_t)(b * CC + c)) * (HH * WWID);
  float Ia = base[y0 * WWID + x0];
  float Ib = base[y1 * WWID + x0];
  float Ic = base[y0 * WWID + x1];
  float Id = base[y1 * WWID + x1];

  zi[(size_t)r * CC + c] = wa * Ia + wb * Ib + wc * Ic + wd * Id;
}

// ---------------------------------------------------------------------------
// K2: h = z_interval @ W1   [2048,256] x [256,256], fp32 WMMA 16x16x4
// one wave per 16x16 output tile; 8 waves / block; 256 blocks
// ---------------------------------------------------------------------------
__global__ void k_gemm1(const float* __restrict__ A,   // z_interval
                        const float* __restrict__ Bm,  // W1
                        float* __restrict__ H) {
  int lane  = threadIdx.x & 31;
  int wave  = threadIdx.x >> 5;
  int tile  = blockIdx.x * 8 + wave;      // 2048 tiles = 128 x 16
  int tm    = tile >> 4;
  int tn    = tile & 15;
  int m     = tm * 16 + (lane & 15);
  int n     = tn * 16 + (lane & 15);
  int khalf = (lane >> 4) * 2;

  v8f acc = {};
  for (int k = 0; k < FF1; k += 4) {
    int ka = k + khalf;
    v2f a = *(const v2f*)(A + (size_t)m * FF1 + ka);
    v2f b;
    b.x = Bm[(size_t)(ka + 0) * FF1 + n];
    b.y = Bm[(size_t)(ka + 1) * FF1 + n];
    acc = __builtin_amdgcn_wmma_f32_16x16x4_f32(false, a, false, b,
                                                (short)0, acc, false, false);
  }
  int rbase = tm * 16 + ((lane >> 4) << 3);
  int col   = tn * 16 + (lane & 15);
#pragma unroll
  for (int v = 0; v < 8; ++v)
    H[(size_t)(rbase + v) * FF1 + col] = acc[v];
}

// ---------------------------------------------------------------------------
// K3: BN train stats per column -> scale/shift
// ---------------------------------------------------------------------------
__global__ void k_bnstats(const float* __restrict__ H,
                          const float* __restrict__ gamma,
                          const float* __restrict__ beta,
                          float* __restrict__ scale,
                          float* __restrict__ shift) {
  int col = blockIdx.x;
  int t   = threadIdx.x;
  float s1 = 0.f, s2 = 0.f;
  for (int r = t; r < N2; r += 256) {
    float v = H[(size_t)r * FF1 + col];
    s1 += v;
    s2 += v * v;
  }
  __shared__ float sh1[256];
  __shared__ float sh2[256];
  sh1[t] = s1; sh2[t] = s2;
  __syncthreads();
  for (int off = 128; off > 0; off >>= 1) {
    if (t < off) { sh1[t] += sh1[t + off]; sh2[t] += sh2[t + off]; }
    __syncthreads();
  }
  if (t == 0) {
    float mean = sh1[0] * (1.0f / (float)N2);
    float var  = sh2[0] * (1.0f / (float)N2) - mean * mean;
    float sc   = gamma[col] / sqrtf(var + 1e-5f);
    scale[col] = sc;
    shift[col] = beta[col] - mean * sc;
  }
}

// ---------------------------------------------------------------------------
// K4: z_pre = relu(bn(h)) @ W2 -- BN+ReLU fused into A-fragment load
// ---------------------------------------------------------------------------
__global__ void k_gemm2(const float* __restrict__ H,
                        const float* __restrict__ scale,
                        const float* __restrict__ shift,
                        const float* __restrict__ W2m,
                        float* __restrict__ ZP) {
  int lane  = threadIdx.x & 31;
  int wave  = threadIdx.x >> 5;
  int tile  = blockIdx.x * 8 + wave;
  int tm    = tile >> 4;
  int tn    = tile & 15;
  int m     = tm * 16 + (lane & 15);
  int n     = tn * 16 + (lane & 15);
  int khalf = (lane >> 4) * 2;

  v8f acc = {};
  for (int k = 0; k < FF1; k += 4) {
    int ka = k + khalf;
    v2f hv = *(const v2f*)(H + (size_t)m * FF1 + ka);
    v2f a;
    a.x = fmaxf(hv.x * scale[ka + 0] + shift[ka + 0], 0.0f);
    a.y = fmaxf(hv.y * scale[ka + 1] + shift[ka + 1], 0.0f);
    v2f b;
    b.x = W2m[(size_t)(ka + 0) * FF1 + n];
    b.y = W2m[(size_t)(ka + 1) * FF1 + n];
    acc = __builtin_amdgcn_wmma_f32_16x16x4_f32(false, a, false, b,
                                                (short)0, acc, false, false);
  }
  int rbase = tm * 16 + ((lane >> 4) << 3);
  int col   = tn * 16 + (lane & 15);
#pragma unroll
  for (int v = 0; v < 8; ++v)
    ZP[(size_t)(rbase + v) * FF1 + col] = acc[v];
}

// ---------------------------------------------------------------------------
// K5: row L2 normalize -> z_box (workspace copy + output copy)
// ---------------------------------------------------------------------------
__global__ void k_norm(const float* __restrict__ ZP,
                       float* __restrict__ ZB,
                       float* __restrict__ out_zbox) {
  int r = blockIdx.x, t = threadIdx.x;
  float v = ZP[(size_t)r * FF1 + t];
  __shared__ float sh[256];
  sh[t] = v * v;
  __syncthreads();
  for (int off = 128; off > 0; off >>= 1) {
    if (t < off) sh[t] += sh[t + off];
    __syncthreads();
  }
  float inv = 1.0f / fmaxf(sqrtf(sh[0]), 1e-12f);
  float z = v * inv;
  ZB[(size_t)r * FF1 + t] = z;
  out_zbox[(size_t)r * FF1 + t] = z;
}

// ---------------------------------------------------------------------------
// K6: s = z_box @ z_box^T with fused diag-drop / pos-col scatter
// point_pos[r]   = s[r, r^1]
// point_neg[r,j] = s[r, j + 2*(j >= (r&~1))]
// ---------------------------------------------------------------------------
__global__ void k_gram(const float* __restrict__ ZB,
                       float* __restrict__ pos,
                       float* __restrict__ neg) {
  int lane  = threadIdx.x & 31;
  int wave  = threadIdx.x >> 5;
  int tile  = blockIdx.x * 8 + wave;      // 16384 tiles = 128 x 128
  int tm    = tile >> 7;
  int tn    = tile & 127;
  int m     = tm * 16 + (lane & 15);
  int n     = tn * 16 + (lane & 15);
  int khalf = (lane >> 4) * 2;

  v8f acc = {};
  for (int k = 0; k < FF1; k += 4) {
    int ka = k + khalf;
    v2f a = *(const v2f*)(ZB + (size_t)m * FF1 + ka);
    v2f b = *(const v2f*)(ZB + (size_t)n * FF1 + ka);  // B[k][n] = ZB[n][k]
    acc = __builtin_amdgcn_wmma_f32_16x16x4_f32(false, a, false, b,
                                                (short)0, acc, false, false);
  }
  int rbase = tm * 16 + ((lane >> 4) << 3);
  int cidx  = tn * 16 + (lane & 15);
#pragma unroll
  for (int v = 0; v < 8; ++v) {
    int   r   = rbase + v;
    float val = acc[v];
    int   p   = r & ~1;
    if (cidx == (r ^ 1)) {
      pos[r] = val;
    } else if (cidx != r) {
      int j = cidx - (cidx > p ? 2 : 0);
      neg[(size_t)r * NNEG + j] = val;
    }
  }
}

// ---------------------------------------------------------------------------
// launch
// ---------------------------------------------------------------------------
extern "C" void kernel_launch(void* const* d_in, const int* in_sizes, int n_in,
                              void* d_out, int out_size, void* d_ws, size_t ws_size,
                              hipStream_t stream) {
  const float* bev     = (const float*)d_in[0];  // [8,256,188,188]
  const float* centers = (const float*)d_in[1];  // [8,256,3]
  const float* W1      = (const float*)d_in[2];  // [256,256]
  const float* gamma   = (const float*)d_in[3];  // [256]
  const float* beta    = (const float*)d_in[4];  // [256]
  const float* W2      = (const float*)d_in[5];  // [256,256]

  float* ws = (float*)d_ws;
  const size_t MAT = (size_t)N2 * FF1;           // 524288 floats
  float* zi    = ws;                              // z_interval
  float* H     = ws + MAT;                        // pre-BN hidden
  float* ZP    = ws + 2 * MAT;                    // pre-norm z_box
  float* ZB    = ws + 3 * MAT;                    // normalized z_box
  float* scale = ws + 4 * MAT;                    // [256]
  float* shift = ws + 4 * MAT + FF1;              // [256]

  float* out = (float*)d_out;
  float* pos      = out;                          // [2048]
  float* negm     = out + N2;                     // [2048, 2046]
  float* out_zbox = out + N2 + (size_t)N2 * NNEG; // [2048, 256]

  k_gather <<<BB * GG, 256, 0, stream>>>(bev, centers, zi);
  k_gemm1  <<<256,     256, 0, stream>>>(zi, W1, H);
  k_bnstats<<<FF1,     256, 0, stream>>>(H, gamma, beta, scale, shift);
  k_gemm2  <<<256,     256, 0, stream>>>(H, scale, shift, W2, ZP);
  k_norm   <<<N2,      256, 0, stream>>>(ZP, ZB, out_zbox);
  k_gram   <<<2048,    256, 0, stream>>>(ZB, pos, negm);
}